// simple_GNN_36893769073203
// MI455X (gfx1250) — compile-verified
//
#include <hip/hip_runtime.h>
#include <math.h>

typedef __attribute__((ext_vector_type(2))) float v2f;
typedef __attribute__((ext_vector_type(8))) float v8f;

#define BN_EPS 1e-5f

// ---------------- BatchNorm statistics: one block per feature ----------------
__global__ void bn_stats_kernel(const float* __restrict__ x, float* __restrict__ stats,
                                int N, int F) {
    int f = blockIdx.x;
    float s = 0.f, s2 = 0.f;
    for (int i = threadIdx.x; i < N; i += blockDim.x) {
        float v = x[(size_t)i * F + f];
        s += v; s2 += v * v;
    }
    __shared__ float sh[256], sh2[256];
    sh[threadIdx.x] = s; sh2[threadIdx.x] = s2;
    __syncthreads();
    for (int off = 128; off > 0; off >>= 1) {
        if ((int)threadIdx.x < off) {
            sh[threadIdx.x]  += sh[threadIdx.x + off];
            sh2[threadIdx.x] += sh2[threadIdx.x + off];
        }
        __syncthreads();
    }
    if (threadIdx.x == 0) {
        float mean = sh[0] / (float)N;
        float var  = sh2[0] / (float)N - mean * mean;
        stats[f]     = mean;
        stats[F + f] = rsqrtf(var + BN_EPS);
    }
}

// ---------------- Apply BN (affine) ----------------
__global__ void bn_apply_kernel(const float* __restrict__ x, const float* __restrict__ stats,
                                const float* __restrict__ gamma, const float* __restrict__ beta,
                                float* __restrict__ xn, int N, int F) {
    int idx = blockIdx.x * blockDim.x + threadIdx.x;
    if (idx >= N * F) return;
    int f = idx & (F - 1);
    xn[idx] = (x[idx] - stats[f]) * stats[F + f] * gamma[f] + beta[f];
}

// ---------------- Degree: init to 1 (self loop), accumulate, rsqrt ----------------
__global__ void deg_init_kernel(float* __restrict__ deg, int N) {
    int i = blockIdx.x * blockDim.x + threadIdx.x;
    if (i < N) deg[i] = 1.0f;
}
__global__ void deg_accum_kernel(const long long* __restrict__ dst, float* __restrict__ deg,
                                 int E) {
    int e = blockIdx.x * blockDim.x + threadIdx.x;
    if (e < E) atomicAdd(&deg[(int)dst[e]], 1.0f);
}
__global__ void deg_rsqrt_kernel(float* __restrict__ deg, int N) {
    int i = blockIdx.x * blockDim.x + threadIdx.x;
    if (i < N) deg[i] = rsqrtf(deg[i]);
}

// ---------------- WMMA f32 GEMM: C[M x 64] = A[M x K] @ B[K x 64] ----------------
// Row-major. M % 16 == 0, K % 4 == 0, Nf fixed at 64.
// One wave computes a 16x64 row-block with 4 accumulators; A loaded once per
// k-step and shared by 4 back-to-back V_WMMA_F32_16X16X4_F32 (independent
// accumulators -> no WMMA->WMMA RAW hazard).
__global__ void gemm_wmma_f32_row_kernel(const float* __restrict__ A, const float* __restrict__ B,
                                         float* __restrict__ C, int M, int K) {
    const int Nf = 64;
    int wave = (blockIdx.x * blockDim.x + threadIdx.x) >> 5;
    int lane = threadIdx.x & 31;
    if (wave * 16 >= M) return;   // wave-uniform exit (EXEC all-1s for WMMA)

    int m0    = wave << 4;
    int l15   = lane & 15;
    int khalf = (lane >> 4) << 1;      // 0 for lanes 0-15, 2 for lanes 16-31
    int mrow  = m0 + l15;

    v8f c0 = {}, c1 = {}, c2 = {}, c3 = {};
    for (int k0 = 0; k0 < K; k0 += 4) {
        // A 16x4 tile: lane holds A[mrow][khalf..khalf+1]
        const float* ap = A + (size_t)mrow * K + k0 + khalf;
        v2f a; a.x = ap[0]; a.y = ap[1];
        // B 4x64 slab: lane holds B[khalf..khalf+1][l15 + 16*t] for tile t
        const float* bp = B + (size_t)(k0 + khalf) * Nf + l15;
        v2f b0, b1, b2, b3;
        b0.x = bp[0];  b0.y = bp[Nf];
        b1.x = bp[16]; b1.y = bp[Nf + 16];
        b2.x = bp[32]; b2.y = bp[Nf + 32];
        b3.x = bp[48]; b3.y = bp[Nf + 48];
        c0 = __builtin_amdgcn_wmma_f32_16x16x4_f32(false, a, false, b0, (short)0, c0, false, false);
        c1 = __builtin_amdgcn_wmma_f32_16x16x4_f32(false, a, false, b1, (short)0, c1, false, false);
        c2 = __builtin_amdgcn_wmma_f32_16x16x4_f32(false, a, false, b2, (short)0, c2, false, false);
        c3 = __builtin_amdgcn_wmma_f32_16x16x4_f32(false, a, false, b3, (short)0, c3, false, false);
    }
    // C/D layout: VGPR v -> row m0 + v + 8*(lane>=16); col = l15 + 16*t
    int mbase = m0 + ((lane >> 4) << 3);
#pragma unroll
    for (int v = 0; v < 8; ++v) {
        float* cp = C + (size_t)(mbase + v) * Nf + l15;
        cp[0]  = c0[v];
        cp[16] = c1[v];
        cp[32] = c2[v];
        cp[48] = c3[v];
    }
}

// ---------------- Zero buffer ----------------
__global__ void zero_kernel(float* __restrict__ p, int n) {
    int i = blockIdx.x * blockDim.x + threadIdx.x;
    if (i < n) p[i] = 0.f;
}

// ---------------- Edge scatter: agg[dst] += h[src] * dinv[src]*dinv[dst] ----------------
// 16 threads per edge, float4 per thread (F = 64).
__global__ void scatter_kernel(const long long* __restrict__ src, const long long* __restrict__ dst,
                               const float* __restrict__ h, const float* __restrict__ dinv,
                               float* __restrict__ agg, int E, int F) {
    long long tid = (long long)blockIdx.x * blockDim.x + threadIdx.x;
    long long e = tid >> 4;
    if (e >= E) return;
    int f4 = (int)(tid & 15) << 2;
    int s = (int)src[e];
    int d = (int)dst[e];
    float norm = dinv[s] * dinv[d];
    const float4 hv = *(const float4*)(h + (size_t)s * F + f4);
    float* out = agg + (size_t)d * F + f4;
    atomicAdd(out + 0, hv.x * norm);
    atomicAdd(out + 1, hv.y * norm);
    atomicAdd(out + 2, hv.z * norm);
    atomicAdd(out + 3, hv.w * norm);
}

// ---------------- Finalize conv: out = relu(agg + h*dinv^2 + bias) ----------------
__global__ void finalize_kernel(const float* __restrict__ agg, const float* __restrict__ h,
                                const float* __restrict__ dinv, const float* __restrict__ bias,
                                float* __restrict__ out, int N, int F) {
    int idx = blockIdx.x * blockDim.x + threadIdx.x;
    if (idx >= N * F) return;
    int i = idx >> 6;          // F == 64
    int f = idx & 63;
    float di = dinv[i];
    float v = agg[idx] + h[idx] * di * di + bias[f];
    out[idx] = v > 0.f ? v : 0.f;
}

// ---------------- Mean pool over nodes: one block per feature ----------------
__global__ void pool_kernel(const float* __restrict__ x, float* __restrict__ pooled,
                            int N, int F) {
    int f = blockIdx.x;
    float s = 0.f;
    for (int i = threadIdx.x; i < N; i += blockDim.x)
        s += x[(size_t)i * F + f];
    __shared__ float sh[256];
    sh[threadIdx.x] = s;
    __syncthreads();
    for (int off = 128; off > 0; off >>= 1) {
        if ((int)threadIdx.x < off) sh[threadIdx.x] += sh[threadIdx.x + off];
        __syncthreads();
    }
    if (threadIdx.x == 0) pooled[f] = sh[0] / (float)N;
}

// ---------------- Head: logits = pooled @ Wlin + blin, softmax ----------------
__global__ void head_kernel(const float* __restrict__ pooled, const float* __restrict__ Wlin,
                            const float* __restrict__ blin, float* __restrict__ out,
                            int F, int FO) {
    if (threadIdx.x == 0) {
        float logits[8];
        float mx = -1e30f;
        for (int o = 0; o < FO; ++o) {
            float s = blin[o];
            for (int f = 0; f < F; ++f) s += pooled[f] * Wlin[f * FO + o];
            logits[o] = s;
            if (s > mx) mx = s;
        }
        float den = 0.f;
        for (int o = 0; o < FO; ++o) { logits[o] = expf(logits[o] - mx); den += logits[o]; }
        for (int o = 0; o < FO; ++o) out[o] = logits[o] / den;
    }
}

static inline int ceil_div(long long a, long long b) { return (int)((a + b - 1) / b); }

extern "C" void kernel_launch(void* const* d_in, const int* in_sizes, int n_in,
                              void* d_out, int out_size, void* d_ws, size_t ws_size,
                              hipStream_t stream) {
    const float* node_feats = (const float*)d_in[0];
    const long long* edge_index = (const long long*)d_in[1];
    const float* bn_gamma = (const float*)d_in[2];
    const float* bn_beta  = (const float*)d_in[3];
    const float* W1 = (const float*)d_in[4];
    const float* b1 = (const float*)d_in[5];
    const float* W2 = (const float*)d_in[6];
    const float* b2 = (const float*)d_in[7];
    const float* W3 = (const float*)d_in[8];
    const float* b3 = (const float*)d_in[9];
    const float* Wlin = (const float*)d_in[10];
    const float* blin = (const float*)d_in[11];

    const int B  = 2;
    const int F  = in_sizes[2];                    // 32
    const int F1 = in_sizes[5];                    // 64
    const int N  = in_sizes[0] / (B * F);          // 50000
    const int E  = in_sizes[1] / (B * 2);          // 1600000
    const int FO = out_size / B;                   // 2

    // workspace layout (floats)
    float* ws     = (float*)d_ws;
    float* stats  = ws;                   // 2*F  (<=64)
    float* pooled = stats + 64;           // F1
    float* dinv   = pooled + 64;          // N
    float* xn     = dinv + N;             // N*F
    float* h      = xn + (size_t)N * F;   // N*F1
    float* agg    = h  + (size_t)N * F1;  // N*F1
    float* xa     = agg + (size_t)N * F1; // N*F1
    float* xb     = xa + (size_t)N * F1;  // N*F1

    for (int b = 0; b < B; ++b) {
        const float* x = node_feats + (size_t)b * N * F;
        const long long* src = edge_index + (size_t)b * 2 * E;
        const long long* dst = src + E;
        float* out = (float*)d_out + (size_t)b * FO;

        // BatchNorm
        bn_stats_kernel<<<F, 256, 0, stream>>>(x, stats, N, F);
        bn_apply_kernel<<<ceil_div((long long)N * F, 256), 256, 0, stream>>>(
            x, stats, bn_gamma, bn_beta, xn, N, F);

        // Degrees -> dinv
        deg_init_kernel<<<ceil_div(N, 256), 256, 0, stream>>>(dinv, N);
        deg_accum_kernel<<<ceil_div(E, 256), 256, 0, stream>>>(dst, dinv, E);
        deg_rsqrt_kernel<<<ceil_div(N, 256), 256, 0, stream>>>(dinv, N);

        const float* convW[3] = {W1, W2, W3};
        const float* convB[3] = {b1, b2, b3};
        const float* xin = xn;
        float* xout_bufs[3] = {xa, xb, xa};
        int Kin = F;

        for (int c = 0; c < 3; ++c) {
            // h = xin @ Wc  (WMMA, one wave per 16x64 row-block)
            int waves = N / 16;
            int gemm_blocks = ceil_div((long long)waves * 32, 256);
            gemm_wmma_f32_row_kernel<<<gemm_blocks, 256, 0, stream>>>(xin, convW[c], h, N, Kin);

            // agg = 0; scatter edges; finalize
            zero_kernel<<<ceil_div((long long)N * F1, 256), 256, 0, stream>>>(agg, N * F1);
            scatter_kernel<<<ceil_div((long long)E * 16, 256), 256, 0, stream>>>(
                src, dst, h, dinv, agg, E, F1);
            finalize_kernel<<<ceil_div((long long)N * F1, 256), 256, 0, stream>>>(
                agg, h, dinv, convB[c], xout_bufs[c], N, F1);

            xin = xout_bufs[c];
            Kin = F1;
        }

        // Mean pool + head
        pool_kernel<<<F1, 256, 0, stream>>>(xin, pooled, N, F1);
        head_kernel<<<1, 32, 0, stream>>>(pooled, Wlin, blin, out, F1, FO);
    }
}